// PairwiseCEFocalLoss_23390391894536
// MI455X (gfx1250) — compile-verified
//
#include <hip/hip_runtime.h>
#include <hip/hip_bf16.h>

// PairwiseCEFocalLoss for B=256, S=512 on gfx1250 (wave32).
// One workgroup per batch row; scores + pos/neg weights staged in LDS;
// V_WMMA_F32_16X16X4_F32 generates 16x16 tiles of d[i,j] = s_j - s_i in f32;
// raw v_exp_f32/v_log_f32 (base-2) focal evaluation per C/D element;
// deterministic reductions.

typedef __attribute__((ext_vector_type(2))) float v2f;
typedef __attribute__((ext_vector_type(8))) float v8f;

#define S_LEN   512
#define TILES   32          // S_LEN / 16
#define NWAVES  8           // 256 threads / wave32
#define ALPHA_C 1.0f
#define SMOOTH_C 1e-7f

__device__ __forceinline__ float fast_exp2(float x) {
#if __has_builtin(__builtin_amdgcn_exp2f)
    return __builtin_amdgcn_exp2f(x);   // raw v_exp_f32, no range fixup
#else
    return __expf(x * 0.6931471805599453f);
#endif
}

__device__ __forceinline__ float fast_log2(float x) {
#if __has_builtin(__builtin_amdgcn_logf)
    return __builtin_amdgcn_logf(x);    // raw v_log_f32 (log base 2)
#else
    return __logf(x) * 1.4426950408889634f;
#endif
}

__global__ __launch_bounds__(256) void pairfocal_rows_kernel(
    const float* __restrict__ scores,
    const int*   __restrict__ targets,
    const int*   __restrict__ tlen,
    float*       __restrict__ per_sample)
{
    __shared__ float sS[S_LEN];   // scores row
    __shared__ float sP[S_LEN];   // pos weight (0/1)
    __shared__ float sN[S_LEN];   // neg weight (0/1)
    __shared__ float waveSum[NWAVES];
    __shared__ int   sNpos, sNneg;

    const int b    = blockIdx.x;
    const int tid  = threadIdx.x;
    const int base = b * S_LEN;

    if (tid == 0) { sNpos = 0; sNneg = 0; }
    __syncthreads();

    // ---- stage row into LDS, build masks, count pos/neg ----
    int cp = 0, cn = 0;
    for (int s = tid; s < S_LEN; s += 256) {
        float sc = scores[base + s];
        int   t  = targets[base + s];
        int   l  = tlen[base + s];
        bool  valid = (l != 0);
        float pw = (valid && t >= 1) ? 1.0f : 0.0f;
        float nw = (valid && t == 0) ? 1.0f : 0.0f;
        sS[s] = sc; sP[s] = pw; sN[s] = nw;
        cp += (pw != 0.0f);
        cn += (nw != 0.0f);
    }
    atomicAdd(&sNpos, cp);   // integer adds: order-independent, deterministic
    atomicAdd(&sNneg, cn);
    __syncthreads();

    const int  wave  = tid >> 5;
    const int  lane  = tid & 31;
    const int  laneN = lane & 15;
    const bool hiH   = (lane >= 16);

    const float LOG2E = 1.4426950408889634f;   // log2(e)
    const float LN2   = 0.6931471805599453f;   // ln(2)

    float acc = 0.0f;

    // Each wave owns pos-tiles ti = wave, wave+8, ... ; full sweep over neg-tiles tj.
    for (int ti = wave; ti < TILES; ti += NWAVES) {
        // A (16x4 f32): A[m,0] = -s_i(m), A[m,1] = 1, A[m,2..3] = 0.
        // Lanes 0-15 hold K=0 (VGPR0), K=1 (VGPR1) of row M=lane; lanes 16-31 -> K=2,3 = 0.
        // Branchless: every lane loads at (lane&15), hi lanes select 0.
        float svA = sS[ti * 16 + laneN];
        v2f a;
        a.x = hiH ? 0.0f : -svA;
        a.y = hiH ? 0.0f : 1.0f;

        // pos weights for this tile's 8 rows per lane (C/D row mapping M = r + 8*hi)
        float pw[8];
#pragma unroll
        for (int r = 0; r < 8; ++r) pw[r] = sP[ti * 16 + r + (hiH ? 8 : 0)];

        for (int tj = 0; tj < TILES; ++tj) {
            // B (4x16 f32): B[0,n] = 1, B[1,n] = s_j(n), B[2..3,n] = 0.
            // VGPR0: lanes0-15 -> K=0, lanes16-31 -> K=2; VGPR1: K=1 / K=3.
            float svB = sS[tj * 16 + laneN];
            v2f bb;
            bb.x = hiH ? 0.0f : 1.0f;
            bb.y = hiH ? 0.0f : svB;

            v8f c = {};
#if __has_builtin(__builtin_amdgcn_wmma_f32_16x16x4_f32)
            // D[m,n] = s_j(n) - s_i(m), full f32 tile in 8 VGPRs/lane
            c = __builtin_amdgcn_wmma_f32_16x16x4_f32(
                    /*neg_a=*/false, a, /*neg_b=*/false, bb,
                    /*c_mod=*/(short)0, c, /*reuse_a=*/false, /*reuse_b=*/false);
#else
#pragma unroll
            for (int r = 0; r < 8; ++r)
                c[r] = svB - sS[ti * 16 + r + (hiH ? 8 : 0)];
#endif
            float nw = sN[tj * 16 + laneN];
#pragma unroll
            for (int r = 0; r < 8; ++r) {
                float d  = c[r];
                float w  = pw[r] * nw;
                // softplus(d) = max(d,0) + ln2 * log2(1 + 2^(-|d|*log2e))
                // exp2 arg <= 0 -> result in (0,1]; log2 arg in (1,2]: raw ops safe.
                float t2 = fast_exp2(-fabsf(d) * LOG2E);
                float sp = fmaxf(d, 0.0f) + LN2 * fast_log2(1.0f + t2);
                // pt = clip(exp(-sp), SMOOTH, 1-SMOOTH)
                float pt = fast_exp2(-sp * LOG2E);
                pt = fminf(fmaxf(pt, SMOOTH_C), 1.0f - SMOOTH_C);
                float omp = 1.0f - pt;
                // pair_loss = -ALPHA * (1-pt)^2 * logpt = ALPHA * (1-pt)^2 * sp
                acc = fmaf(w, ALPHA_C * omp * omp * sp, acc);
            }
        }
    }

    // ---- deterministic reductions: wave32 butterfly, then fixed-order wave sum ----
#pragma unroll
    for (int off = 16; off >= 1; off >>= 1)
        acc += __shfl_xor(acc, off, 32);
    if (lane == 0) waveSum[wave] = acc;
    __syncthreads();

    if (tid == 0) {
        float tot = 0.0f;
#pragma unroll
        for (int w = 0; w < NWAVES; ++w) tot += waveSum[w];
        int cnt = sNpos * sNneg;   // pair count factorizes
        per_sample[b] = (cnt > 0) ? (tot / (float)cnt) : 0.0f;
    }
}

__global__ __launch_bounds__(256) void pairfocal_final_kernel(
    const float* __restrict__ per_sample, float* __restrict__ out, int B)
{
    __shared__ float sm[256];
    int tid = threadIdx.x;
    sm[tid] = (tid < B) ? per_sample[tid] : 0.0f;
    __syncthreads();
    for (int s = 128; s > 0; s >>= 1) {
        if (tid < s) sm[tid] += sm[tid + s];
        __syncthreads();
    }
    if (tid == 0) out[0] = sm[0] / (float)B;
}

extern "C" void kernel_launch(void* const* d_in, const int* in_sizes, int n_in,
                              void* d_out, int out_size, void* d_ws, size_t ws_size,
                              hipStream_t stream) {
    const float* scores  = (const float*)d_in[0];
    const int*   targets = (const int*)d_in[1];
    const int*   tlen    = (const int*)d_in[2];
    float* out = (float*)d_out;

    const int B = in_sizes[0] / S_LEN;   // 256 for the reference shapes
    float* per_sample = (float*)d_ws;    // B floats of scratch

    pairfocal_rows_kernel<<<B, 256, 0, stream>>>(scores, targets, tlen, per_sample);
    pairfocal_final_kernel<<<1, 256, 0, stream>>>(per_sample, out, B);
}